// GCNLayer_32298154066114
// MI455X (gfx1250) — compile-verified
//
#include <hip/hip_runtime.h>

#define N_NODES 8192
#define N_EDGES 262144
#define DIN     128
#define DOUT    128

typedef float v2f __attribute__((ext_vector_type(2)));
typedef float v8f __attribute__((ext_vector_type(8)));

// -------- Kernel 1: h = x @ W^T + b  via V_WMMA_F32_16X16X4_F32 --------
// One block = 16 rows of x (staged in LDS), 8 waves each own a 16-col tile of h.
__global__ void __launch_bounds__(256)
gemm_bias_kernel(const float* __restrict__ x,
                 const float* __restrict__ W,
                 const float* __restrict__ bias,
                 float* __restrict__ h) {
    __shared__ float As[16 * DIN];            // 8 KB x-tile
    const int tid  = threadIdx.x;
    const int wave = tid >> 5;
    const int lane = tid & 31;
    const int m0   = blockIdx.x * 16;
    const int n0   = wave * 16;

    // cooperative load: 2048 floats / 256 threads = 8 each
#pragma unroll
    for (int i = 0; i < 8; ++i) {
        const int off = i * 256 + tid;
        As[off] = x[m0 * DIN + off];
    }
    __syncthreads();

    const int half = lane >> 4;               // 0: lanes 0-15, 1: lanes 16-31
    const int l16  = lane & 15;
    const int ncol = n0 + l16;                // B/C/D column for this lane

    v8f acc = {};
    for (int kk = 0; kk < DIN; kk += 4) {
        const int ka = kk + half * 2;         // K = {0,1} (lanes 0-15) / {2,3} (16-31)
        v2f a, b;
        // A (16x4, M x K): lane holds row M=l16, two consecutive K values
        a.x = As[l16 * DIN + ka];
        a.y = As[l16 * DIN + ka + 1];
        // B (4x16, K x N): B[k][n] = W[n][k]  (h = x @ W^T)
        b.x = W[ncol * DIN + ka];
        b.y = W[ncol * DIN + ka + 1];
        // 8-arg form: (neg_a, A, neg_b, B, c_mod, C, reuse_a, reuse_b)
        acc = __builtin_amdgcn_wmma_f32_16x16x4_f32(
            false, a, false, b, (short)0, acc, false, false);
    }

    // C/D layout: VGPR j -> M = j + 8*half, N = l16
    const float bv = bias[ncol];
#pragma unroll
    for (int j = 0; j < 8; ++j) {
        const int m = m0 + j + half * 8;
        h[m * DOUT + ncol] = acc[j] + bv;
    }
}

// -------- Kernel 2: deg[row[e]] += 1 --------
__global__ void degree_kernel(const int* __restrict__ row,
                              float* __restrict__ deg) {
    const int e = blockIdx.x * blockDim.x + threadIdx.x;
    if (e < N_EDGES) atomicAdd(&deg[row[e]], 1.0f);
}

// -------- Kernel 3: dinv = deg > 0 ? deg^-0.5 : 0 --------
__global__ void dinv_kernel(const float* __restrict__ deg,
                            float* __restrict__ dinv) {
    const int i = blockIdx.x * blockDim.x + threadIdx.x;
    if (i < N_NODES) {
        const float d = deg[i];
        dinv[i] = (d > 0.0f) ? rsqrtf(d) : 0.0f;
    }
}

// -------- Kernel 4: out[r,:] += norm * h[c,:] (one wave per edge, deduped) --------
__global__ void __launch_bounds__(256)
aggregate_kernel(const int* __restrict__ ei,
                 const float* __restrict__ dinv,
                 const float* __restrict__ h,
                 unsigned long long* __restrict__ bitmap,   // may be null
                 float* __restrict__ out) {
    const int gtid = blockIdx.x * blockDim.x + threadIdx.x;
    const int e    = gtid >> 5;
    const int lane = gtid & 31;
    if (e >= N_EDGES) return;

    const int r = ei[e];
    const int c = ei[N_EDGES + e];

    // Set-semantics dedup: only the first occurrence of (r,c) contributes.
    int keep = 1;
    if (bitmap) {
        int k0 = 0;
        if (lane == 0) {
            const long long idx = (long long)r * N_NODES + c;
            const unsigned long long m = 1ull << (idx & 63);
            const unsigned long long old = atomicOr(&bitmap[idx >> 6], m);
            k0 = ((old & m) == 0) ? 1 : 0;
        }
        keep = __shfl(k0, 0, 32);
    }
    if (!keep) return;

    const float norm = dinv[r] * dinv[c];
#pragma unroll
    for (int j = 0; j < 4; ++j) {
        const int col = lane + j * 32;
        atomicAdd(&out[r * DOUT + col], norm * h[c * DOUT + col]);
    }
}

// -------- Kernel 5: in-place ReLU --------
__global__ void relu_kernel(float* __restrict__ out) {
    const int i = blockIdx.x * blockDim.x + threadIdx.x;
    if (i < N_NODES * DOUT) out[i] = fmaxf(out[i], 0.0f);
}

extern "C" void kernel_launch(void* const* d_in, const int* in_sizes, int n_in,
                              void* d_out, int out_size, void* d_ws, size_t ws_size,
                              hipStream_t stream) {
    const float* x  = (const float*)d_in[0];          // [N, DIN]
    const int*   ei = (const int*)d_in[1];            // [2, E]
    const float* W  = (const float*)d_in[2];          // [DOUT, DIN]
    const float* b  = (const float*)d_in[3];          // [DOUT]
    float* out      = (float*)d_out;                  // [N, DOUT] f32

    // Workspace layout (bytes)
    const size_t h_bytes   = (size_t)N_NODES * DOUT * sizeof(float);      // 4 MB
    const size_t deg_bytes = (size_t)N_NODES * sizeof(float);             // 32 KB
    const size_t bm_bytes  = ((size_t)N_NODES * N_NODES) / 8;             // 8 MB
    char* ws = (char*)d_ws;
    float* h    = (float*)ws;
    float* deg  = (float*)(ws + h_bytes);
    float* dinv = (float*)(ws + h_bytes + deg_bytes);
    const size_t bm_off = h_bytes + 2 * deg_bytes;
    const bool dedup = (ws_size >= bm_off + bm_bytes);
    unsigned long long* bitmap = dedup ? (unsigned long long*)(ws + bm_off) : nullptr;

    // Zero accumulators (graph-capturable memset nodes)
    hipMemsetAsync(deg, 0, deg_bytes, stream);
    if (dedup) hipMemsetAsync(bitmap, 0, bm_bytes, stream);
    hipMemsetAsync(out, 0, (size_t)N_NODES * DOUT * sizeof(float), stream);

    // 1) Linear (WMMA f32)
    gemm_bias_kernel<<<N_NODES / 16, 256, 0, stream>>>(x, W, b, h);
    // 2) Degree
    degree_kernel<<<(N_EDGES + 255) / 256, 256, 0, stream>>>(ei, deg);
    // 3) deg^-0.5
    dinv_kernel<<<(N_NODES + 255) / 256, 256, 0, stream>>>(deg, dinv);
    // 4) Normalized aggregation (one wave per edge)
    aggregate_kernel<<<(N_EDGES * 32) / 256, 256, 0, stream>>>(ei, dinv, h, bitmap, out);
    // 5) ReLU
    relu_kernel<<<(N_NODES * DOUT) / 256, 256, 0, stream>>>(out);
}